// PTDNetGCN_53008486367827
// MI455X (gfx1250) — compile-verified
//
#include <hip/hip_runtime.h>
#include <cstddef>

#define GAMMA_C (-0.1f)
#define ZETA_C  (1.1f)

typedef float v2f __attribute__((ext_vector_type(2)));
typedef float v8f __attribute__((ext_vector_type(8)));

// ---------------------------------------------------------------------------
// h = X[N,128] @ W[128,Dout]   (fp32 WMMA 16x16x4, one wave per 16x16 tile)
// A layout (16x4 f32): lanes 0-15 -> M=lane, v0/v1 = K=0/1 ; lanes 16-31 -> K=2/3
// B layout (4x16 f32): lanes 0-15 -> N=lane, v0/v1 = K=0/1 ; lanes 16-31 -> K=2/3
// C/D (16x16 f32): vgpr r, lane<16 -> D[r][lane], lane>=16 -> D[r+8][lane-16]
// ---------------------------------------------------------------------------
__global__ void gemm_xw_wmma(const float* __restrict__ X,
                             const float* __restrict__ W,
                             float* __restrict__ H,
                             int N, int Dout) {
  const int K = 128;
  const int tile_m = blockIdx.x;
  const int tile_n = blockIdx.y * blockDim.y + threadIdx.y;
  const int lane = threadIdx.x;           // 0..31, full wave
  const int hsel = lane >> 4;             // 0 or 1  (selects K pair 0/1 vs 2/3)
  const int l16  = lane & 15;

  const int arow = tile_m * 16 + l16;     // A-matrix row held by this lane
  const int bcol = tile_n * 16 + l16;     // B-matrix col held by this lane

  const float* __restrict__ xr = X + (size_t)arow * K;

  v8f c = {};
#pragma unroll 8
  for (int k = 0; k < K; k += 4) {
    const int ka = k + hsel * 2;
    v2f a, b;
    a.x = xr[ka];
    a.y = xr[ka + 1];
    b.x = W[(size_t)ka * Dout + bcol];
    b.y = W[(size_t)(ka + 1) * Dout + bcol];
    c = __builtin_amdgcn_wmma_f32_16x16x4_f32(
            /*neg_a=*/false, a, /*neg_b=*/false, b,
            /*c_mod=*/(short)0, c, /*reuse_a=*/false, /*reuse_b=*/false);
  }

  const int rbase = tile_m * 16 + hsel * 8;
  const int ccol  = tile_n * 16 + l16;
#pragma unroll
  for (int r = 0; r < 8; ++r) {
    H[(size_t)(rbase + r) * Dout + ccol] = c[r];
  }
}

// ---------------------------------------------------------------------------
// Per-node attention scalars:
//   a[n] = relu(x[n]@nb_w  + nb_b) . att_w[0:16]
//   b[n] = relu(x[n]@self_w+ self_b). att_w[16:32]
// ---------------------------------------------------------------------------
__global__ void node_att(const float* __restrict__ x,
                         const float* __restrict__ nbw, const float* __restrict__ nbb,
                         const float* __restrict__ sw,  const float* __restrict__ sb,
                         const float* __restrict__ attw,
                         float* __restrict__ av, float* __restrict__ bv, int N) {
  const int K = 128, H1 = 16;
  __shared__ float s_nbw[128 * 16];
  __shared__ float s_sw[128 * 16];
  __shared__ float s_attw[32];
  __shared__ float s_nbb[16];
  __shared__ float s_sb[16];
  for (int i = threadIdx.x; i < K * H1; i += blockDim.x) {
    s_nbw[i] = nbw[i];
    s_sw[i]  = sw[i];
  }
  if (threadIdx.x < 32) s_attw[threadIdx.x] = attw[threadIdx.x];
  if (threadIdx.x < 16) { s_nbb[threadIdx.x] = nbb[threadIdx.x]; s_sb[threadIdx.x] = sb[threadIdx.x]; }
  __syncthreads();

  const int n = blockIdx.x * blockDim.x + threadIdx.x;
  if (n >= N) return;
  const float* __restrict__ xr = x + (size_t)n * K;

  float p[16], q[16];
#pragma unroll
  for (int j = 0; j < 16; ++j) { p[j] = s_nbb[j]; q[j] = s_sb[j]; }
  for (int k = 0; k < K; ++k) {
    const float xv = xr[k];
#pragma unroll
    for (int j = 0; j < 16; ++j) {
      p[j] = fmaf(xv, s_nbw[k * 16 + j], p[j]);
      q[j] = fmaf(xv, s_sw[k * 16 + j],  q[j]);
    }
  }
  float aa = 0.f, bb = 0.f;
#pragma unroll
  for (int j = 0; j < 16; ++j) {
    aa = fmaf(fmaxf(p[j], 0.f), s_attw[j],      aa);
    bb = fmaf(fmaxf(q[j], 0.f), s_attw[16 + j], bb);
  }
  av[n] = aa;
  bv[n] = bb;
}

// rowsum[n] = 1.0  (self-loop value)
__global__ void fill1(float* __restrict__ p, int n) {
  int i = blockIdx.x * blockDim.x + threadIdx.x;
  if (i < n) p[i] = 1.0f;
}

// mask[e] = clip(sigmoid(a[row]+b[col]+att_b)*(zeta-gamma)+gamma, 0, 1); rowsum[row] += mask
__global__ void edge_mask(const int* __restrict__ row, const int* __restrict__ col,
                          const float* __restrict__ av, const float* __restrict__ bv,
                          const float* __restrict__ attb,
                          float* __restrict__ mask, float* __restrict__ rowsum, int E) {
  int e = blockIdx.x * blockDim.x + threadIdx.x;
  if (e >= E) return;
  const int r = row[e], cc = col[e];
  const float logit = av[r] + bv[cc] + attb[0];
  const float gate = 1.0f / (1.0f + expf(-logit));
  float m = gate * (ZETA_C - GAMMA_C) + GAMMA_C;
  m = fminf(fmaxf(m, 0.0f), 1.0f);
  mask[e] = m;
  if (m != 0.0f) atomicAdd(&rowsum[r], m);
}

// dis[n] = clip(rowsum[n]^-0.5, 0, 10)   (rowsum >= 1, so rsqrt is safe)
__global__ void dinv_k(const float* __restrict__ rs, float* __restrict__ dis, int n) {
  int i = blockIdx.x * blockDim.x + threadIdx.x;
  if (i < n) dis[i] = fminf(rsqrtf(rs[i]), 10.0f);
}

// out[n,c] = dis[n]^2 * h[n,c]   (the identity/self-loop contribution)
__global__ void out_init(const float* __restrict__ h, const float* __restrict__ dis,
                         float* __restrict__ out, long long total, int Dout) {
  long long i = (long long)blockIdx.x * blockDim.x + threadIdx.x;
  if (i >= total) return;
  const int n = (int)(i / Dout);
  const float d = dis[n];
  out[i] = d * d * h[i];
}

// per-edge aggregation: out[row] += mask*dis[row]*dis[col] * h[col]
// one wave (32 lanes) per edge, Dout/32 elements per lane
__global__ void edge_agg(const int* __restrict__ row, const int* __restrict__ col,
                         const float* __restrict__ mask, const float* __restrict__ dis,
                         const float* __restrict__ h, float* __restrict__ out,
                         int E, int Dout) {
  const long long gid = (long long)blockIdx.x * blockDim.x + threadIdx.x;
  const int lane = (int)(gid & 31);
  const long long e = gid >> 5;
  if (e >= E) return;
  const int r = row[e], cc = col[e];
  const float cf = mask[e] * dis[r] * dis[cc];
  if (cf == 0.0f) return;           // uniform across the wave (same edge)
  const float* __restrict__ hc = h + (size_t)cc * Dout;
  float* __restrict__ o = out + (size_t)r * Dout;
  for (int k = lane; k < Dout; k += 32) {
    atomicAdd(&o[k], cf * hc[k]);
  }
}

__global__ void relu_k(float* __restrict__ p, long long n) {
  long long i = (long long)blockIdx.x * blockDim.x + threadIdx.x;
  if (i < n) p[i] = fmaxf(p[i], 0.0f);
}

// ---------------------------------------------------------------------------

extern "C" void kernel_launch(void* const* d_in, const int* in_sizes, int n_in,
                              void* d_out, int out_size, void* d_ws, size_t ws_size,
                              hipStream_t stream) {
  const float* x      = (const float*)d_in[0];
  const int*   row    = (const int*)  d_in[1];
  const int*   col    = (const int*)  d_in[2];
  const float* nb_w   = (const float*)d_in[3];
  const float* nb_b   = (const float*)d_in[4];
  const float* self_w = (const float*)d_in[5];
  const float* self_b = (const float*)d_in[6];
  const float* att_w  = (const float*)d_in[7];
  const float* att_b  = (const float*)d_in[8];
  const float* w0     = (const float*)d_in[9];
  const float* w1     = (const float*)d_in[10];
  float* out = (float*)d_out;

  const int D   = 128;
  const int N   = in_sizes[0] / D;     // 100000
  const int E   = in_sizes[1];         // 1600000
  const int HID = 128;
  const int OUT = 64;

  // workspace layout (fp32)
  char* ws = (char*)d_ws;
  const size_t szH = (size_t)N * 128 * sizeof(float);   // 51.2 MB
  const size_t szN = (size_t)N * sizeof(float);
  float* h      = (float*)(ws);
  float* acc    = (float*)(ws + szH);
  float* av     = (float*)(ws + 2 * szH);
  float* bv     = (float*)(ws + 2 * szH + szN);
  float* rowsum = (float*)(ws + 2 * szH + 2 * szN);
  float* dis    = (float*)(ws + 2 * szH + 3 * szN);
  float* mask   = (float*)(ws + 2 * szH + 4 * szN);

  const int TB = 256;
  const int nBlkN   = (N + TB - 1) / TB;
  const int nBlkE   = (E + TB - 1) / TB;
  const int nBlkE32 = (int)(((long long)E * 32 + TB - 1) / TB);

  for (int layer = 0; layer < 2; ++layer) {
    const float* xin   = (layer == 0) ? x   : acc;
    const float* wg    = (layer == 0) ? w0  : w1;
    const int    dout  = (layer == 0) ? HID : OUT;
    float*       dst   = (layer == 0) ? acc : out;
    const float* l_nbw = nb_w   + (size_t)layer * D * 16;
    const float* l_nbb = nb_b   + (size_t)layer * 16;
    const float* l_sw  = self_w + (size_t)layer * D * 16;
    const float* l_sb  = self_b + (size_t)layer * 16;
    const float* l_aw  = att_w  + (size_t)layer * 32;
    const float* l_ab  = att_b  + (size_t)layer * 1;

    // 1) h = xin @ wg   (WMMA, one wave per 16x16 tile, 4 waves per block)
    {
      dim3 grid(N / 16, dout / 64);
      dim3 block(32, 4);
      gemm_xw_wmma<<<grid, block, 0, stream>>>(xin, wg, h, N, dout);
    }
    // 2) per-node attention scalars
    node_att<<<nBlkN, TB, 0, stream>>>(xin, l_nbw, l_nbb, l_sw, l_sb, l_aw, av, bv, N);
    // 3) rowsum = 1 (self loop)
    fill1<<<nBlkN, TB, 0, stream>>>(rowsum, N);
    // 4) edge masks + degree accumulation
    edge_mask<<<nBlkE, TB, 0, stream>>>(row, col, av, bv, l_ab, mask, rowsum, E);
    // 5) d^{-1/2}
    dinv_k<<<nBlkN, TB, 0, stream>>>(rowsum, dis, N);
    // 6) diagonal (self-loop) term
    {
      long long total = (long long)N * dout;
      int blk = (int)((total + TB - 1) / TB);
      out_init<<<blk, TB, 0, stream>>>(h, dis, dst, total, dout);
    }
    // 7) edge aggregation (wave per edge)
    edge_agg<<<nBlkE32, TB, 0, stream>>>(row, col, mask, dis, h, dst, E, dout);
    // 8) activation (relu on layer 0 only; layer 1 is identity)
    if (layer == 0) {
      long long total = (long long)N * dout;
      int blk = (int)((total + TB - 1) / TB);
      relu_k<<<blk, TB, 0, stream>>>(dst, total);
    }
  }
}